// TransD_27831388078834
// MI455X (gfx1250) — compile-verified
//
#include <hip/hip_runtime.h>

// ---------------------------------------------------------------------------
// TransD margin loss, MI455X (gfx1250, wave32).
// Gather-bandwidth bound: ~1.07 GB of random 512B entity-row gathers,
// floor ~46us @ 23.3 TB/s. One wave per (pos,neg) pair; dim 128 = 32 lanes x
// float4. Entity rows are staged LDS-side with the CDNA5 async-to-LDS engine
// (global_load_async_to_lds_b128, s_wait_asynccnt) in a 2-deep double-buffered
// pipeline; reductions use ds_swizzle xor-trees; deterministic 2-stage final
// reduction (no float atomics).
// ---------------------------------------------------------------------------

constexpr int   kEntDim        = 128;
constexpr int   kBatch         = 262144;
constexpr float kMargin        = 1.0f;
constexpr float kEps           = 1e-6f;
constexpr int   kBlocks        = 4096;
constexpr int   kThreads       = 256;                       // 8 waves/block
constexpr int   kWavesPerBlock = kThreads / 32;
constexpr int   kTotalWaves    = kBlocks * kWavesPerBlock;  // 32768 -> 8 pairs/wave

// Types matching the builtin's declared parameters (from hipcc diagnostic):
// param0: generic pointer to int __attribute__((vector_size(16)))
// param1: LDS pointer to same vector type
typedef int v4i_vs __attribute__((vector_size(16)));
typedef __attribute__((address_space(3))) v4i_vs* as3_v4i_ptr;

#if __has_builtin(__builtin_amdgcn_global_load_async_to_lds_b128)
#define USE_ASYNC_LDS 1
#else
#define USE_ASYNC_LDS 0
#endif

#if __has_builtin(__builtin_amdgcn_s_wait_asynccnt)
#define WAIT_ASYNC(N)                                   \
  do {                                                  \
    __builtin_amdgcn_s_wait_asynccnt(N);                \
    asm volatile("" ::: "memory");                      \
  } while (0)
#else
#define WAIT_ASYNC(N) asm volatile("s_wait_asynccnt %0" ::"i"(N) : "memory")
#endif

// wave32 xor-tree reduction via ds_swizzle immediates (group-of-32 xor mode:
// offset = xor_mask<<10 | and_mask(0x1f)). 5 ds ops, no lane-address math.
__device__ __forceinline__ float wave_sum32(float v) {
  v += __int_as_float(__builtin_amdgcn_ds_swizzle(__float_as_int(v), (16 << 10) | 0x1f));
  v += __int_as_float(__builtin_amdgcn_ds_swizzle(__float_as_int(v), (8 << 10) | 0x1f));
  v += __int_as_float(__builtin_amdgcn_ds_swizzle(__float_as_int(v), (4 << 10) | 0x1f));
  v += __int_as_float(__builtin_amdgcn_ds_swizzle(__float_as_int(v), (2 << 10) | 0x1f));
  v += __int_as_float(__builtin_amdgcn_ds_swizzle(__float_as_int(v), (1 << 10) | 0x1f));
  return v;  // all 32 lanes hold the full sum
}

// Shared score math: d = (relp*dh + head) + rel - (relp*dt + tail) + EPS
__device__ __forceinline__ float score_math(float4 head, float4 headp,
                                            float4 tail, float4 tailp,
                                            float4 rl, float4 rp) {
  float dh = head.x * headp.x + head.y * headp.y + head.z * headp.z + head.w * headp.w;
  float dt = tail.x * tailp.x + tail.y * tailp.y + tail.z * tailp.z + tail.w * tailp.w;
  dh = wave_sum32(dh);
  dt = wave_sum32(dt);
  const float dx = fmaf(rp.x, dh, head.x) + rl.x - fmaf(rp.x, dt, tail.x) + kEps;
  const float dy = fmaf(rp.y, dh, head.y) + rl.y - fmaf(rp.y, dt, tail.y) + kEps;
  const float dz = fmaf(rp.z, dh, head.z) + rl.z - fmaf(rp.z, dt, tail.z) + kEps;
  const float dw = fmaf(rp.w, dh, head.w) + rl.w - fmaf(rp.w, dt, tail.w) + kEps;
  float ss = dx * dx + dy * dy + dz * dz + dw * dw;
  ss = wave_sum32(ss);
  return sqrtf(ss);
}

#if USE_ASYNC_LDS
// Issue one 512B entity row: lane l moves bytes [16l,16l+16) global -> LDS.
__device__ __forceinline__ void async_row(const float* gsrc, float* ldst, int c) {
  __builtin_amdgcn_global_load_async_to_lds_b128(
      (v4i_vs*)(gsrc + c), (as3_v4i_ptr)(ldst + c), 0, 0);
}

// 8 rows per (pos,neg) pair: pos{head,headp,tail,tailp}, neg{...}
__device__ __forceinline__ void stage_pair(const float* ent, const float* entp,
                                           int ph, int pt, int nh, int nt,
                                           float* buf, int c) {
  async_row(ent  + (size_t)ph * kEntDim, buf + 0 * kEntDim, c);
  async_row(entp + (size_t)ph * kEntDim, buf + 1 * kEntDim, c);
  async_row(ent  + (size_t)pt * kEntDim, buf + 2 * kEntDim, c);
  async_row(entp + (size_t)pt * kEntDim, buf + 3 * kEntDim, c);
  async_row(ent  + (size_t)nh * kEntDim, buf + 4 * kEntDim, c);
  async_row(entp + (size_t)nh * kEntDim, buf + 5 * kEntDim, c);
  async_row(ent  + (size_t)nt * kEntDim, buf + 6 * kEntDim, c);
  async_row(entp + (size_t)nt * kEntDim, buf + 7 * kEntDim, c);
}

__device__ __forceinline__ float score_from_lds(const float* buf, float4 rl,
                                                float4 rp, int c) {
  const float4 head  = *(const float4*)(buf + 0 * kEntDim + c);
  const float4 headp = *(const float4*)(buf + 1 * kEntDim + c);
  const float4 tail  = *(const float4*)(buf + 2 * kEntDim + c);
  const float4 tailp = *(const float4*)(buf + 3 * kEntDim + c);
  return score_math(head, headp, tail, tailp, rl, rp);
}
#else
// Fallback: direct per-lane b128 gathers into VGPRs.
__device__ __forceinline__ float score_direct(const float* __restrict__ ent,
                                              const float* __restrict__ entp,
                                              const float* __restrict__ rel,
                                              const float* __restrict__ relp,
                                              int h, int r, int t, int c) {
  const float4 head  = *(const float4*)(ent  + (size_t)h * kEntDim + c);
  const float4 headp = *(const float4*)(entp + (size_t)h * kEntDim + c);
  const float4 tail  = *(const float4*)(ent  + (size_t)t * kEntDim + c);
  const float4 tailp = *(const float4*)(entp + (size_t)t * kEntDim + c);
  const float4 rl    = *(const float4*)(rel  + (size_t)r * kEntDim + c);
  const float4 rp    = *(const float4*)(relp + (size_t)r * kEntDim + c);
  return score_math(head, headp, tail, tailp, rl, rp);
}
#endif

__global__ __launch_bounds__(kThreads) void transd_partial_kernel(
    const float* __restrict__ ent, const float* __restrict__ entp,
    const float* __restrict__ rel, const float* __restrict__ relp,
    const int* __restrict__ pos, const int* __restrict__ neg,
    float* __restrict__ partial) {
  const int lane = threadIdx.x & 31;
  const int wave = threadIdx.x >> 5;
  const int wid  = blockIdx.x * kWavesPerBlock + wave;
  const int c    = lane * 4;

  float acc = 0.0f;

#if USE_ASYNC_LDS
  // Double-buffered stage: 8 waves x 2 bufs x 8 rows x 512B = 64 KB/block.
  __shared__ __align__(16) float stage[kWavesPerBlock * 2 * 8 * kEntDim];
  float* const wbuf = stage + wave * (2 * 8 * kEntDim);

  int i  = wid;
  int ph = pos[3 * i + 0], pr = pos[3 * i + 1], pt = pos[3 * i + 2];
  int nh = neg[3 * i + 0], nr = neg[3 * i + 1], nt = neg[3 * i + 2];
  stage_pair(ent, entp, ph, pt, nh, nt, wbuf, c);  // prologue -> buf 0
  int parity = 0;

  while (i < kBatch) {
    const int  inext = i + kTotalWaves;
    const bool more  = inext < kBatch;  // uniform across all waves
    int ph2 = 0, pr2 = 0, pt2 = 0, nh2 = 0, nr2 = 0, nt2 = 0;
    if (more) {
      ph2 = pos[3 * inext + 0]; pr2 = pos[3 * inext + 1]; pt2 = pos[3 * inext + 2];
      nh2 = neg[3 * inext + 0]; nr2 = neg[3 * inext + 1]; nt2 = neg[3 * inext + 2];
      stage_pair(ent, entp, ph2, pt2, nh2, nt2,
                 wbuf + (parity ^ 1) * 8 * kEntDim, c);
    }
    // rel tables are 512KB -> L2-resident; plain VGPR loads.
    const float4 prl = *(const float4*)(rel  + (size_t)pr * kEntDim + c);
    const float4 prp = *(const float4*)(relp + (size_t)pr * kEntDim + c);
    const float4 nrl = *(const float4*)(rel  + (size_t)nr * kEntDim + c);
    const float4 nrp = *(const float4*)(relp + (size_t)nr * kEntDim + c);

    // Async loads complete in order: <=8 outstanding => current batch landed.
    if (more) WAIT_ASYNC(8); else WAIT_ASYNC(0);

    const float* cur = wbuf + parity * 8 * kEntDim;
    const float  ps  = score_from_lds(cur, prl, prp, c);
    const float  ns  = score_from_lds(cur + 4 * kEntDim, nrl, nrp, c);
    const float  l   = ps - ns + kMargin;
    acc += (l > 0.0f) ? l : 0.0f;

    ph = ph2; pr = pr2; pt = pt2; nh = nh2; nr = nr2; nt = nt2;
    parity ^= 1;
    i = inext;
  }
#else
  for (int i = wid; i < kBatch; i += kTotalWaves) {
    const int ph = pos[3 * i + 0], pr = pos[3 * i + 1], pt = pos[3 * i + 2];
    const int nh = neg[3 * i + 0], nr = neg[3 * i + 1], nt = neg[3 * i + 2];
    const float ps = score_direct(ent, entp, rel, relp, ph, pr, pt, c);
    const float ns = score_direct(ent, entp, rel, relp, nh, nr, nt, c);
    const float l  = ps - ns + kMargin;
    acc += (l > 0.0f) ? l : 0.0f;
  }
#endif

  __shared__ float ssum[kWavesPerBlock];
  if (lane == 0) ssum[wave] = acc;
  __syncthreads();
  if (threadIdx.x == 0) {
    float s = 0.0f;
#pragma unroll
    for (int j = 0; j < kWavesPerBlock; ++j) s += ssum[j];
    partial[blockIdx.x] = s;
  }
}

// Deterministic final reduction of kBlocks partials -> scalar loss.
__global__ __launch_bounds__(256) void transd_reduce_kernel(
    const float* __restrict__ partial, int n, float* __restrict__ out) {
  __shared__ float smem[256];
  float acc = 0.0f;
  for (int idx = threadIdx.x; idx < n; idx += 256) acc += partial[idx];
  smem[threadIdx.x] = acc;
  __syncthreads();
#pragma unroll
  for (int off = 128; off > 0; off >>= 1) {
    if (threadIdx.x < (unsigned)off) smem[threadIdx.x] += smem[threadIdx.x + off];
    __syncthreads();
  }
  if (threadIdx.x == 0) out[0] = smem[0] * (1.0f / (float)kBatch);
}

extern "C" void kernel_launch(void* const* d_in, const int* in_sizes, int n_in,
                              void* d_out, int out_size, void* d_ws, size_t ws_size,
                              hipStream_t stream) {
  const float* ent  = (const float*)d_in[0];  // ent_emb     [1e6,128] f32
  const float* entp = (const float*)d_in[1];  // ent_map_emb [1e6,128] f32
  const float* rel  = (const float*)d_in[2];  // rel_emb     [1e3,128] f32
  const float* relp = (const float*)d_in[3];  // rel_map_emb [1e3,128] f32
  const int*   pos  = (const int*)d_in[4];    // pos_x [262144,3] i32
  const int*   neg  = (const int*)d_in[5];    // neg_x [262144,3] i32

  float* partial = (float*)d_ws;              // kBlocks floats (16 KB)

  transd_partial_kernel<<<dim3(kBlocks), dim3(kThreads), 0, stream>>>(
      ent, entp, rel, relp, pos, neg, partial);
  transd_reduce_kernel<<<dim3(1), dim3(256), 0, stream>>>(
      partial, kBlocks, (float*)d_out);
}